// GAT_37460704755886
// MI455X (gfx1250) — compile-verified
//
#include <hip/hip_runtime.h>
#include <math.h>

typedef __attribute__((ext_vector_type(2))) float v2f;
typedef __attribute__((ext_vector_type(8))) float v8f;

#define GAT_NN   50000
#define GAT_NE   800000
#define GAT_FIN  128
#define GAT_H1   4
#define GAT_D1   64
#define GAT_C1   256   /* H1*D1 */
#define GAT_C2   128
#define GAT_SLOPE 0.2f

// ---------------------------------------------------------------------------
// Grid-stride f32 fill (for -inf / 0 init of accumulators; d_ws is poisoned).
// ---------------------------------------------------------------------------
__global__ void gat_init_f32(float* __restrict__ p, float v, long n) {
    long t = (long)blockIdx.x * blockDim.x + threadIdx.x;
    long stride = (long)gridDim.x * blockDim.x;
    for (; t < n; t += stride) p[t] = v;
}

// ---------------------------------------------------------------------------
// FP32 GEMM C[M,N] = A[M,K] @ B[K,N] using V_WMMA_F32_16X16X4_F32.
// Register-blocked: one wave computes a 16x64 strip = 4 WMMA accumulators,
// reusing one A fragment across 4 back-to-back WMMAs per K-step (D->C
// accumulate chains carry no hazard NOPs; only D->A/B does).
// Requires M % 16 == 0, N % 64 == 0, K % 4 == 0 (exact here), so EXEC is
// all-1s at every WMMA (early-out below is wave-uniform).
// A frag (16x4): lane m = lane&15; vgpr pair holds K={0,1} (lanes 0-15),
// K={2,3} (lanes 16-31). B frag (4x16) symmetric. C/D: vgpr r -> row r /
// r+8 by lane half, col = lane&15.
// ---------------------------------------------------------------------------
__global__ __launch_bounds__(256)
void gat_gemm_wmma_f32(const float* __restrict__ A, const float* __restrict__ B,
                       float* __restrict__ C, int M, int N, int K) {
    int wid  = (int)((blockIdx.x * blockDim.x + threadIdx.x) >> 5);
    int lane = threadIdx.x & 31;
    int groupsN = N >> 6;                    // 64-wide column groups
    int tm = wid / groupsN;
    int tg = wid % groupsN;
    if (tm * 16 >= M) return;                // wave-uniform; EXEC stays full

    const float* Arow = A + (size_t)tm * 16 * K;
    const float* Bcol = B + tg * 64;
    int m    = lane & 15;
    int koff = (lane >> 4) << 1;             // 0 for lanes 0-15, 2 for lanes 16-31

    v8f acc[4] = {};
    for (int k = 0; k < K; k += 4) {
        v2f a;
        a.x = Arow[(size_t)m * K + (k + koff)];
        a.y = Arow[(size_t)m * K + (k + koff + 1)];
        const float* b0 = Bcol + (size_t)(k + koff) * N;       // row k+koff
        const float* b1 = Bcol + (size_t)(k + koff + 1) * N;   // row k+koff+1
#pragma unroll
        for (int sub = 0; sub < 4; ++sub) {
            v2f b;
            b.x = b0[sub * 16 + m];
            b.y = b1[sub * 16 + m];
            acc[sub] = __builtin_amdgcn_wmma_f32_16x16x4_f32(
                    /*neg_a=*/false, a, /*neg_b=*/false, b,
                    /*c_mod=*/(short)0, acc[sub],
                    /*reuse_a=*/false, /*reuse_b=*/false);
        }
    }

    int row0 = (lane >> 4) << 3;             // 0 or 8
    int col  = lane & 15;
    float* Cbase = C + (size_t)tm * 16 * N + tg * 64;
#pragma unroll
    for (int sub = 0; sub < 4; ++sub)
#pragma unroll
        for (int r = 0; r < 8; ++r)
            Cbase[(size_t)(row0 + r) * N + sub * 16 + col] = acc[sub][r];
}

// ---------------------------------------------------------------------------
// Per-(node,head) attention dots: el[n,h] = <feat[n,h,:], al[h,:]>, er likewise.
// One wave per (n,h); lane-strided loads, shfl-xor reduction (wave32).
// ---------------------------------------------------------------------------
__global__ void gat_node_dots(const float* __restrict__ feat,
                              const float* __restrict__ al,
                              const float* __restrict__ ar,
                              float* __restrict__ el, float* __restrict__ er,
                              int Nn, int H, int D) {
    int wid  = (int)((blockIdx.x * blockDim.x + threadIdx.x) >> 5);
    int lane = threadIdx.x & 31;
    if (wid >= Nn * H) return;
    int n = wid / H, h = wid % H;
    const float* f = feat + (size_t)n * H * D + h * D;
    float sl = 0.f, sr = 0.f;
    for (int d = lane; d < D; d += 32) {
        float v = f[d];
        sl += v * al[h * D + d];
        sr += v * ar[h * D + d];
    }
#pragma unroll
    for (int off = 16; off > 0; off >>= 1) {
        sl += __shfl_xor(sl, off, 32);
        sr += __shfl_xor(sr, off, 32);
    }
    if (lane == 0) { el[wid] = sl; er[wid] = sr; }
}

__device__ __forceinline__ float gat_lrelu(float v) {
    return v > 0.f ? v : GAT_SLOPE * v;
}

// ---------------------------------------------------------------------------
// Pass 1: segment max over dst (float atomic max via int bit-trick).
// m must be initialized to -inf.
// ---------------------------------------------------------------------------
__global__ void gat_edge_max(const int* __restrict__ src, const int* __restrict__ dst,
                             const float* __restrict__ el, const float* __restrict__ er,
                             float* __restrict__ m, int E, int H) {
    long t = (long)blockIdx.x * blockDim.x + threadIdx.x;
    if (t >= (long)E * H) return;
    int e = (int)(t / H), h = (int)(t % H);
    float v = gat_lrelu(el[src[e] * H + h] + er[dst[e] * H + h]);
    float* addr = &m[dst[e] * H + h];
    if (v >= 0.f) atomicMax((int*)addr, __float_as_int(v));
    else          atomicMin((unsigned int*)addr, __float_as_uint(v));
}

// ---------------------------------------------------------------------------
// Pass 2: denom[dst,h] += exp(e - m[dst,h])
// ---------------------------------------------------------------------------
__global__ void gat_edge_expsum(const int* __restrict__ src, const int* __restrict__ dst,
                                const float* __restrict__ el, const float* __restrict__ er,
                                const float* __restrict__ m, float* __restrict__ denom,
                                int E, int H) {
    long t = (long)blockIdx.x * blockDim.x + threadIdx.x;
    if (t >= (long)E * H) return;
    int e = (int)(t / H), h = (int)(t % H);
    int d = dst[e];
    float v = gat_lrelu(el[src[e] * H + h] + er[d * H + h]);
    atomicAdd(&denom[d * H + h], expf(v - m[d * H + h]));
}

// ---------------------------------------------------------------------------
// Pass 3: out[dst,:] += alpha * feat[src,:]. H*D/4 threads per edge,
// float4 gathers + 4 atomic f32 adds per thread.
// ---------------------------------------------------------------------------
__global__ void gat_edge_aggregate(const int* __restrict__ src, const int* __restrict__ dst,
                                   const float* __restrict__ el, const float* __restrict__ er,
                                   const float* __restrict__ m, const float* __restrict__ denom,
                                   const float* __restrict__ feat, float* __restrict__ out,
                                   int E, int H, int D) {
    int C = H * D;
    int tpe = C >> 2;                       // threads per edge
    long t = (long)blockIdx.x * blockDim.x + threadIdx.x;
    if (t >= (long)E * tpe) return;
    int e  = (int)(t / tpe);
    int q  = (int)(t % tpe);
    int c0 = q << 2;
    int h  = c0 / D;
    int s = src[e], d = dst[e];
    float v = gat_lrelu(el[s * H + h] + er[d * H + h]);
    float alpha = expf(v - m[d * H + h]) / denom[d * H + h];
    const float4 f = *(const float4*)(feat + (size_t)s * C + c0);
    float* o = out + (size_t)d * C + c0;
    atomicAdd(o + 0, alpha * f.x);
    atomicAdd(o + 1, alpha * f.y);
    atomicAdd(o + 2, alpha * f.z);
    atomicAdd(o + 3, alpha * f.w);
}

// ---------------------------------------------------------------------------
// In-place bias add (+ optional ReLU).
// ---------------------------------------------------------------------------
__global__ void gat_bias_act(float* __restrict__ out, const float* __restrict__ bias,
                             long n, int C, int do_relu) {
    long t = (long)blockIdx.x * blockDim.x + threadIdx.x;
    if (t >= n) return;
    float v = out[t] + bias[(int)(t % C)];
    out[t] = do_relu ? fmaxf(v, 0.f) : v;
}

static inline int gat_cdiv(long a, long b) { return (int)((a + b - 1) / b); }

extern "C" void kernel_launch(void* const* d_in, const int* in_sizes, int n_in,
                              void* d_out, int out_size, void* d_ws, size_t ws_size,
                              hipStream_t stream) {
    (void)in_sizes; (void)n_in; (void)out_size; (void)ws_size;
    const float* x   = (const float*)d_in[0];
    const int*   src = (const int*)  d_in[1];
    const int*   dst = (const int*)  d_in[2];
    const float* W1  = (const float*)d_in[3];
    const float* al1 = (const float*)d_in[4];
    const float* ar1 = (const float*)d_in[5];
    const float* b1  = (const float*)d_in[6];
    const float* W2  = (const float*)d_in[7];
    const float* al2 = (const float*)d_in[8];
    const float* ar2 = (const float*)d_in[9];
    const float* b2  = (const float*)d_in[10];
    float* out = (float*)d_out;

    // Workspace layout (floats): total 50000*660 = 33.0M floats (= 132 MB)
    float* ws   = (float*)d_ws;
    float* h1   = ws;                                   // [N, 256]
    float* out1 = h1   + (size_t)GAT_NN * GAT_C1;       // [N, 256] (becomes relu'd L2 input)
    float* h2   = out1 + (size_t)GAT_NN * GAT_C1;       // [N, 128]
    float* el1  = h2   + (size_t)GAT_NN * GAT_C2;       // [N, 4]
    float* er1  = el1  + (size_t)GAT_NN * GAT_H1;
    float* m1   = er1  + (size_t)GAT_NN * GAT_H1;
    float* dn1  = m1   + (size_t)GAT_NN * GAT_H1;
    float* el2  = dn1  + (size_t)GAT_NN * GAT_H1;       // [N, 1]
    float* er2  = el2  + GAT_NN;
    float* m2   = er2  + GAT_NN;
    float* dn2  = m2   + GAT_NN;

    const int BLK = 256;

    // ---------------- Layer 1 ----------------
    // h1 = x @ W1 : M=50000, N=256, K=128 -> (3125 * 4) waves of 16x64 strips
    {
        long waves = (long)(GAT_NN / 16) * (GAT_C1 / 64);
        gat_gemm_wmma_f32<<<gat_cdiv(waves * 32, BLK), BLK, 0, stream>>>(
            x, W1, h1, GAT_NN, GAT_C1, GAT_FIN);
    }

    // el1/er1
    gat_node_dots<<<gat_cdiv((long)GAT_NN * GAT_H1 * 32, BLK), BLK, 0, stream>>>(
        h1, al1, ar1, el1, er1, GAT_NN, GAT_H1, GAT_D1);

    // init accumulators
    gat_init_f32<<<2048, BLK, 0, stream>>>(m1,  -INFINITY, (long)GAT_NN * GAT_H1);
    gat_init_f32<<<2048, BLK, 0, stream>>>(dn1, 0.f,       (long)GAT_NN * GAT_H1);
    gat_init_f32<<<4096, BLK, 0, stream>>>(out1, 0.f,      (long)GAT_NN * GAT_C1);

    // edge softmax + aggregate
    gat_edge_max<<<gat_cdiv((long)GAT_NE * GAT_H1, BLK), BLK, 0, stream>>>(
        src, dst, el1, er1, m1, GAT_NE, GAT_H1);
    gat_edge_expsum<<<gat_cdiv((long)GAT_NE * GAT_H1, BLK), BLK, 0, stream>>>(
        src, dst, el1, er1, m1, dn1, GAT_NE, GAT_H1);
    gat_edge_aggregate<<<gat_cdiv((long)GAT_NE * (GAT_C1 / 4), BLK), BLK, 0, stream>>>(
        src, dst, el1, er1, m1, dn1, h1, out1, GAT_NE, GAT_H1, GAT_D1);

    // out1 = relu(out1 + b1)  (in-place; becomes layer-2 input)
    gat_bias_act<<<gat_cdiv((long)GAT_NN * GAT_C1, BLK), BLK, 0, stream>>>(
        out1, b1, (long)GAT_NN * GAT_C1, GAT_C1, 1);

    // ---------------- Layer 2 ----------------
    // h2 = out1 @ W2 : M=50000, N=128, K=256 -> (3125 * 2) waves
    {
        long waves = (long)(GAT_NN / 16) * (GAT_C2 / 64);
        gat_gemm_wmma_f32<<<gat_cdiv(waves * 32, BLK), BLK, 0, stream>>>(
            out1, W2, h2, GAT_NN, GAT_C2, GAT_C1);
    }

    gat_node_dots<<<gat_cdiv((long)GAT_NN * 32, BLK), BLK, 0, stream>>>(
        h2, al2, ar2, el2, er2, GAT_NN, 1, GAT_C2);

    gat_init_f32<<<2048, BLK, 0, stream>>>(m2,  -INFINITY, (long)GAT_NN);
    gat_init_f32<<<2048, BLK, 0, stream>>>(dn2, 0.f,       (long)GAT_NN);
    gat_init_f32<<<4096, BLK, 0, stream>>>(out, 0.f,       (long)GAT_NN * GAT_C2);

    gat_edge_max<<<gat_cdiv((long)GAT_NE, BLK), BLK, 0, stream>>>(
        src, dst, el2, er2, m2, GAT_NE, 1);
    gat_edge_expsum<<<gat_cdiv((long)GAT_NE, BLK), BLK, 0, stream>>>(
        src, dst, el2, er2, m2, dn2, GAT_NE, 1);
    gat_edge_aggregate<<<gat_cdiv((long)GAT_NE * (GAT_C2 / 4), BLK), BLK, 0, stream>>>(
        src, dst, el2, er2, m2, dn2, h2, out, GAT_NE, 1, GAT_C2);

    // out += b2 (no relu)
    gat_bias_act<<<gat_cdiv((long)GAT_NN * GAT_C2, BLK), BLK, 0, stream>>>(
        out, b2, (long)GAT_NN * GAT_C2, GAT_C2, 0);
}